// batch_Q_approx_51135880626199
// MI455X (gfx1250) — compile-verified
//
#include <hip/hip_runtime.h>

typedef __attribute__((ext_vector_type(2))) float v2f;
typedef __attribute__((ext_vector_type(8))) float v8f;

#define NNETS 256
#define HID 128
#define BATCH 2048
#define ROWS_PER_BLK 128
#define W2_STRIDE 132   // 128 + 4 pad: (4*row + col) % 64 -> conflict-free A-frag reads

__global__ __launch_bounds__(256) void mlp256_fused_wmma_f32(
    const float* __restrict__ a,  const float* __restrict__ W1,
    const float* __restrict__ b1, const float* __restrict__ W2,
    const float* __restrict__ b2, const float* __restrict__ W3,
    const float* __restrict__ b3, float* __restrict__ out)
{
    // Static LDS: 70,144 bytes (gfx1250 WGP has 320 KB; up to 4 blocks/WGP).
    __shared__ float sW2[HID * W2_STRIDE];  // padded row-major W2[kout][h]
    __shared__ float sW1[HID];
    __shared__ float sB1[HID];
    __shared__ float sB2[HID];
    __shared__ float sW3[HID];
    __shared__ float sA[ROWS_PER_BLK];

    const int n  = blockIdx.y;            // net id
    const int r0 = blockIdx.x * ROWS_PER_BLK;
    const int t  = threadIdx.x;

    // ---- stage small per-net vectors + this block's slice of a ----
    if (t < HID) {
        sW1[t] = W1[n * HID + t];
        sB1[t] = b1[n * HID + t];
        sB2[t] = b2[n * HID + t];
        sW3[t] = W3[n * HID + t];
        sA[t]  = a[(size_t)n * BATCH + r0 + t];
    }

    // ---- stage W2 (128x128 f32) into padded LDS, coalesced b64 ----
    const float* gW2 = W2 + (size_t)n * HID * HID;
    #pragma unroll
    for (int i = 0; i < 32; ++i) {
        int idx2 = t + 256 * i;           // float2 index 0..8191
        int e    = idx2 * 2;
        int row  = e >> 7;
        int col  = e & 127;
        v2f v = *(const v2f*)(gW2 + e);
        *(v2f*)(sW2 + row * W2_STRIDE + col) = v;
    }
    __syncthreads();

    const int lane = t & 31;
    const int wv   = t >> 5;              // wave 0..7 -> owns 16 batch rows
    const int li   = lane & 15;
    const int lh   = lane >> 4;           // lane half: K offset 0 / 2

    // ---- layer 1 (rank-1) computed directly into B-fragments in VGPRs ----
    // B-frag layout (4x16 f32): lane%16 = N (batch row), K = vgpr + 2*(lane/16)
    const int brow = wv * 16 + li;        // batch row within block
    const float aval = sA[brow];
    v2f bfr[32];
    #pragma unroll
    for (int s = 0; s < 32; ++s) {
        int k = 4 * s + 2 * lh;
        v2f w1v = *(const v2f*)(sW1 + k);
        v2f b1v = *(const v2f*)(sB1 + k);
        v2f h;
        h.x = fmaxf(aval * w1v.x + b1v.x, 0.0f);
        h.y = fmaxf(aval * w1v.y + b1v.y, 0.0f);
        bfr[s] = h;
    }

    // ---- layer 2 via v_wmma_f32_16x16x4_f32, fused with layer-3 dot ----
    // Compute D = W2(16x4) x h1^T(4x16): M = kout tile, N = batch rows.
    float acc = 0.0f;
    #pragma unroll
    for (int mt = 0; mt < 8; ++mt) {
        v8f c = {0.f, 0.f, 0.f, 0.f, 0.f, 0.f, 0.f, 0.f};
        const float* arow = sW2 + (mt * 16 + li) * W2_STRIDE + 2 * lh;
        #pragma unroll
        for (int s = 0; s < 32; ++s) {
            v2f afrag = *(const v2f*)(arow + 4 * s);   // A: lane%16 = M, K = v + 2*lh
            c = __builtin_amdgcn_wmma_f32_16x16x4_f32(
                    false, afrag, false, bfr[s], (short)0, c, false, false);
        }
        // D layout: c[v] = h2_pre[batch = brow, kout = mt*16 + v + 8*lh]
        #pragma unroll
        for (int v = 0; v < 8; ++v) {
            int kout = mt * 16 + v + 8 * lh;
            float h2 = fmaxf(c[v] + sB2[kout], 0.0f);
            acc += h2 * sW3[kout];
        }
    }

    // ---- reduce the two lane-halves (kout 0..7 vs 8..15 of each tile) ----
    acc += __shfl_xor(acc, 16, 32);
    if (lh == 0) {
        out[(size_t)n * BATCH + r0 + brow] = acc + b3[n];
    }
}

extern "C" void kernel_launch(void* const* d_in, const int* in_sizes, int n_in,
                              void* d_out, int out_size, void* d_ws, size_t ws_size,
                              hipStream_t stream) {
    const float* a  = (const float*)d_in[0];
    const float* W1 = (const float*)d_in[1];
    const float* b1 = (const float*)d_in[2];
    const float* W2 = (const float*)d_in[3];
    const float* b2 = (const float*)d_in[4];
    const float* W3 = (const float*)d_in[5];
    const float* b3 = (const float*)d_in[6];
    float* out = (float*)d_out;

    dim3 grid(BATCH / ROWS_PER_BLK, NNETS);   // 16 x 256 blocks
    mlp256_fused_wmma_f32<<<grid, 256, 0, stream>>>(a, W1, b1, W2, b2, W3, b3, out);
}